// NodeEncoder_31980326486308
// MI455X (gfx1250) — compile-verified
//
#include <hip/hip_runtime.h>
#include <hip/hip_bf16.h>

typedef float v2f __attribute__((ext_vector_type(2)));
typedef float v4f __attribute__((ext_vector_type(4)));
typedef float v8f __attribute__((ext_vector_type(8)));

#define D 128
#define EPS 1e-5f

// ---------------------------------------------------------------------------
// Kernel 1: LayerNorm (no affine). One wave32 per node, 4 f32 per lane.
// node_inputs is read exactly once -> non-temporal load (keep L2 for h/agg).
// ---------------------------------------------------------------------------
__global__ void ln_kernel(const float* __restrict__ x,
                          float* __restrict__ xn, int N) {
  int node = blockIdx.x * (blockDim.x >> 5) + (threadIdx.x >> 5);
  if (node >= N) return;
  int lane = threadIdx.x & 31;
  const v4f* xp = (const v4f*)(x + (size_t)node * D);
  v4f v = __builtin_nontemporal_load(xp + lane);

  float s = v.x + v.y + v.z + v.w;
#pragma unroll
  for (int off = 16; off > 0; off >>= 1) s += __shfl_xor(s, off, 32);
  float mu = s * (1.0f / D);

  float dx = v.x - mu, dy = v.y - mu, dz = v.z - mu, dw = v.w - mu;
  float s2 = dx * dx + dy * dy + dz * dz + dw * dw;
#pragma unroll
  for (int off = 16; off > 0; off >>= 1) s2 += __shfl_xor(s2, off, 32);
  float rstd = __frsqrt_rn(s2 * (1.0f / D) + EPS);

  v4f o;
  o.x = dx * rstd; o.y = dy * rstd; o.z = dz * rstd; o.w = dw * rstd;
  ((v4f*)(xn + (size_t)node * D))[lane] = o;  // RT: re-read by GEMM + final
}

// ---------------------------------------------------------------------------
// Kernel 2: h = xn @ W via V_WMMA_F32_16X16X4_F32 (full fp32, matches ref).
// One wave per 16x16 output tile; K=128 -> 32 chained WMMAs.
// A 16x4 f32 layout: lanes 0-15 row M=lane K={0,1}; lanes 16-31 K={2,3}.
// C/D 16x16 f32 layout: VGPR r -> M=r (lanes 0-15) / M=8+r (lanes 16-31).
// ---------------------------------------------------------------------------
__global__ void gemm_wmma_kernel(const float* __restrict__ xn,
                                 const float* __restrict__ W,
                                 float* __restrict__ h, int N) {
  const int colTiles = D / 16;  // 8
  int tile = blockIdx.x * (blockDim.x >> 5) + (threadIdx.x >> 5);  // wave id
  int numTiles = (N / 16) * colTiles;
  if (tile >= numTiles) return;  // uniform per wave: EXEC stays all-ones

  int rowTile = tile / colTiles;
  int colTile = tile % colTiles;
  int lane  = threadIdx.x & 31;
  int mrow  = rowTile * 16 + (lane & 15);   // A row for this lane
  int ncol  = colTile * 16 + (lane & 15);   // B col for this lane
  int khalf = (lane >> 4) * 2;              // which K pair this lane holds

  v8f acc = {};
#pragma unroll
  for (int k = 0; k < D; k += 4) {
    v2f a, b;
    a.x = xn[(size_t)mrow * D + k + khalf];
    a.y = xn[(size_t)mrow * D + k + khalf + 1];
    b.x = W[(size_t)(k + khalf) * D + ncol];
    b.y = W[(size_t)(k + khalf + 1) * D + ncol];
    acc = __builtin_amdgcn_wmma_f32_16x16x4_f32(
        /*neg_a=*/false, a, /*neg_b=*/false, b,
        /*c_mod=*/(short)0, acc, /*reuse_a=*/false, /*reuse_b=*/false);
  }

  int m0 = (lane < 16) ? 0 : 8;
  int n  = lane & 15;
#pragma unroll
  for (int r = 0; r < 8; ++r) {
    h[(size_t)(rowTile * 16 + m0 + r) * D + colTile * 16 + n] = acc[r];
  }
}

// ---------------------------------------------------------------------------
// Kernel 3: agg = broadcast(b)  (RT stores: agg is the atomic hot set)
// ---------------------------------------------------------------------------
__global__ void init_agg_kernel(float* __restrict__ agg,
                                const float* __restrict__ b, int total) {
  int i = blockIdx.x * blockDim.x + threadIdx.x;
  if (i < total) agg[i] = b[i & (D - 1)];
}

// ---------------------------------------------------------------------------
// Kernel 4: edge scatter. One wave per edge: gather h[src] (L2-resident, RT),
// scale by w_e, atomic-add into agg[dst] (resolves in L2).
// Edge list / weights are streamed once -> non-temporal loads.
// ---------------------------------------------------------------------------
__global__ void edge_kernel(const float* __restrict__ h,
                            const long long* __restrict__ eidx,
                            const float* __restrict__ ew,
                            float* __restrict__ agg, int E) {
  int e = blockIdx.x * (blockDim.x >> 5) + (threadIdx.x >> 5);
  if (e >= E) return;
  int lane = threadIdx.x & 31;
  long long src = __builtin_nontemporal_load(eidx + e);
  long long dst = __builtin_nontemporal_load(eidx + (long long)E + e);
  float w = __builtin_nontemporal_load(ew + e);

  v4f v = ((const v4f*)(h + (size_t)src * D))[lane];  // RT gather (L2 hit)
  float* ap = agg + (size_t)dst * D + lane * 4;
  atomicAdd(ap + 0, v.x * w);
  atomicAdd(ap + 1, v.y * w);
  atomicAdd(ap + 2, v.z * w);
  atomicAdd(ap + 3, v.w * w);
}

// ---------------------------------------------------------------------------
// Kernel 5: out = agg * dropout_mask - xn  (all operands last-use -> NT)
// ---------------------------------------------------------------------------
__global__ void final_kernel(const float* __restrict__ agg,
                             const float* __restrict__ mask,
                             const float* __restrict__ xn,
                             float* __restrict__ out, int total) {
  int i4 = blockIdx.x * blockDim.x + threadIdx.x;  // one v4f per thread
  if (i4 * 4 < total) {
    v4f a = __builtin_nontemporal_load((const v4f*)agg  + i4);
    v4f m = __builtin_nontemporal_load((const v4f*)mask + i4);
    v4f x = __builtin_nontemporal_load((const v4f*)xn   + i4);
    v4f o;
    o.x = a.x * m.x - x.x;
    o.y = a.y * m.y - x.y;
    o.z = a.z * m.z - x.z;
    o.w = a.w * m.w - x.w;
    __builtin_nontemporal_store(o, (v4f*)out + i4);
  }
}

// ---------------------------------------------------------------------------
extern "C" void kernel_launch(void* const* d_in, const int* in_sizes, int n_in,
                              void* d_out, int out_size, void* d_ws, size_t ws_size,
                              hipStream_t stream) {
  const float*     x    = (const float*)d_in[0];      // node_inputs [N, D]
  const long long* eidx = (const long long*)d_in[1];  // edge_index  [2, E] int64
  const float*     ew   = (const float*)d_in[2];      // edge_weights [E]
  const float*     W    = (const float*)d_in[3];      // W [D, D]
  const float*     b    = (const float*)d_in[4];      // b [D]
  const float*     mask = (const float*)d_in[5];      // dropout_mask [N, D]
  float*           out  = (float*)d_out;

  const int N = in_sizes[0] / D;
  const int E = in_sizes[2];
  const int total = N * D;

  float* xn  = (float*)d_ws;          // [N, D] normalized inputs
  float* h   = xn  + (size_t)total;   // [N, D] xn @ W
  float* agg = h   + (size_t)total;   // [N, D] segment sum + b

  // 1) LayerNorm: 8 waves (nodes) per 256-thread block
  {
    int wavesPerBlock = 8;
    int blocks = (N + wavesPerBlock - 1) / wavesPerBlock;
    ln_kernel<<<blocks, 256, 0, stream>>>(x, xn, N);
  }

  // 2) GEMM via fp32 WMMA: one wave per 16x16 tile
  {
    int numTiles = (N / 16) * (D / 16);  // 3125 * 8 = 25000
    int wavesPerBlock = 8;
    int blocks = (numTiles + wavesPerBlock - 1) / wavesPerBlock;
    gemm_wmma_kernel<<<blocks, 256, 0, stream>>>(xn, W, h, N);
  }

  // 3) agg = b (broadcast)
  {
    int blocks = (total + 255) / 256;
    init_agg_kernel<<<blocks, 256, 0, stream>>>(agg, b, total);
  }

  // 4) edge scatter-add: one wave per edge
  {
    int wavesPerBlock = 8;
    int blocks = (E + wavesPerBlock - 1) / wavesPerBlock;
    edge_kernel<<<blocks, 256, 0, stream>>>(h, eidx, ew, agg, E);
  }

  // 5) out = agg * mask - xn  (vectorized: total/4 v4f elements)
  {
    int quads = total / 4;
    int blocks = (quads + 255) / 256;
    final_kernel<<<blocks, 256, 0, stream>>>(agg, mask, xn, out, total);
  }
}